// TwoHotEmbedding_13030930776069
// MI455X (gfx1250) — compile-verified
//
#include <hip/hip_runtime.h>
#include <stdint.h>

// ---------------------------------------------------------------------------
// TwoHotEmbedding on MI455X (gfx1250)
//
// out[r,:] = weight[i1[r],:] + (i1[r]!=i2[r] ? weight[i2[r],:] : 0)
//
// Pure memory-bound gather (~192 MB traffic, ~8 us at 23.3 TB/s; no matmul
// structure -> WMMA inapplicable). Uses the CDNA5 Tensor Data Mover in
// GATHER mode: one tensor_load_to_lds DMAs 8 index-selected rows (256 f32)
// of the embedding table into LDS; two TDMs per wave (one per index set),
// tracked with TENSORcnt. Indices are scalar-loaded (s_load_b256) directly
// into the SGPR octets used as D# groups 2/3. Output stores are non-temporal
// so the 64 MB write stream does not evict the ~102 MB table from the
// 192 MB L2 (the gathers then run at L2 bandwidth, not HBM).
// ---------------------------------------------------------------------------

typedef __attribute__((ext_vector_type(4))) unsigned u32x4;
typedef __attribute__((ext_vector_type(8))) unsigned u32x8;
typedef __attribute__((ext_vector_type(4))) float    f32x4;

#define EMBED_DIM        256
#define NUM_EMB          100000
#define ROWS_PER_GROUP   8          // gather mode: up to 8 rows w/ 32-bit indices
#define WAVES_PER_BLOCK  2
#define THREADS          (WAVES_PER_BLOCK * 32)
// Per wave: 2 regions (A,B) * 8 rows * 256 f32 = 4096 floats = 16 KB.
// Block: 2 waves * 16 KB = 32 KB static LDS -> ~10 blocks resident per WGP.

__global__ __launch_bounds__(THREADS)
void twohot_tdm_kernel(const unsigned* __restrict__ idx1,
                       const unsigned* __restrict__ idx2,
                       const float*    __restrict__ weight,
                       float*          __restrict__ out)
{
    __shared__ float smem[WAVES_PER_BLOCK * 4096];

    const int lane = threadIdx.x & 31;
    // Wave index is uniform per wave but the compiler can't prove it from
    // threadIdx; readfirstlane makes it an SGPR so everything derived from it
    // (index loads, descriptors, LDS addresses) scalarizes.
    const int wave    = __builtin_amdgcn_readfirstlane(threadIdx.x >> 5);
    const int group   = blockIdx.x * WAVES_PER_BLOCK + wave;
    const int rowbase = group * ROWS_PER_GROUP;

    float* lA = &smem[wave * 4096];       // 8 rows gathered via idx1
    float* lB = lA + 2048;                // 8 rows gathered via idx2

    // Low 32 bits of a generic pointer into LDS == LDS byte offset
    // (aperture occupies the high 32 bits; ISA 10.2 aperture calc).
    const unsigned ldsA = (unsigned)(uintptr_t)lA;
    const unsigned ldsB = (unsigned)(uintptr_t)lB;

    // Uniform scalar loads of the 8 row indices per set -> s_load_b256,
    // landing directly in the SGPR octets that become D# groups 2/3.
    u32x8 va, vb;
#pragma unroll
    for (int r = 0; r < ROWS_PER_GROUP; ++r) {
        va[r] = idx1[rowbase + r];
        vb[r] = idx2[rowbase + r];
    }

    // ---- Tensor DMA descriptor (D#), gather mode, 32-bit indices ----------
    const uint64_t ga    = (uint64_t)(uintptr_t)weight;
    const unsigned ga_lo = (unsigned)ga;
    const unsigned ga_hi = ((unsigned)(ga >> 32)) & 0x01FFFFFFu; // addr[56:32]

    // Group 0: [1:0]=count=1, [30]=gather_index_size(32b), [31]=gather_mode,
    //          [63:32]=lds_addr, [120:64]=global_addr, [127:126]=type=2.
    const u32x4 g0a = { 0xC0000001u, ldsA, ga_lo, ga_hi | 0x80000000u };
    const u32x4 g0b = { 0xC0000001u, ldsB, ga_lo, ga_hi | 0x80000000u };

    // Group 1:
    //  d0: workgroup_mask=0 (not in cluster), data_size=2 (4 bytes)
    //  d1: atomic_barrier_addr=0, tensor_dim0[15:0]=256
    //  d2: tensor_dim0[31:16]=0, tensor_dim1[15:0]=100000&0xFFFF=0x86A0
    //  d3: tensor_dim1[31:16]=1, tile_dim0=256
    //  d4: tile_dim1=#valid indices=8 (tile_dim2 ignored in gather mode)
    //  d5: tensor_dim0_stride[31:0]=256 elements
    //  d6,d7: stride hi / tensor_dim1_stride (ignored in gather mode) = 0
    const u32x8 g1 = { 0x00020000u, 0x01000000u, 0x86A00000u, 0x01000001u,
                       0x00000008u, 0x00000100u, 0x00000000u, 0x00000000u };

    // Groups 2/3: the eight 32-bit row indices per set.
    const u32x4 g2a = { va[0], va[1], va[2], va[3] };
    const u32x4 g3a = { va[4], va[5], va[6], va[7] };
    const u32x4 g2b = { vb[0], vb[1], vb[2], vb[3] };
    const u32x4 g3b = { vb[4], vb[5], vb[6], vb[7] };

    // ---- Issue the two gather DMAs (async, TENSORcnt) ---------------------
    asm volatile("tensor_load_to_lds %0, %1, %2, %3"
                 :: "s"(g0a), "s"(g1), "s"(g2a), "s"(g3a) : "memory");
    asm volatile("tensor_load_to_lds %0, %1, %2, %3"
                 :: "s"(g0b), "s"(g1), "s"(g2b), "s"(g3b) : "memory");

    __builtin_amdgcn_s_wait_tensorcnt(0);
    asm volatile("" ::: "memory");   // fence LDS reads behind the wait

    // ---- Combine: 256 f32/row -> 2 x float4 per lane per row --------------
    const f32x4* A4 = (const f32x4*)lA;      // 8 rows * 64 float4, contiguous
    const f32x4* B4 = (const f32x4*)lB;
    f32x4* o4 = (f32x4*)out + (size_t)rowbase * (EMBED_DIM / 4);

#pragma unroll
    for (int r = 0; r < ROWS_PER_GROUP; ++r) {
        const float s = (va[r] == vb[r]) ? 0.0f : 1.0f;  // two-hot collision fix
        f32x4 a0 = A4[r * 64 + lane];
        f32x4 a1 = A4[r * 64 + lane + 32];
        f32x4 b0 = B4[r * 64 + lane];
        f32x4 b1 = B4[r * 64 + lane + 32];
        f32x4 o0 = a0 + s * b0;
        f32x4 o1 = a1 + s * b1;
        // Streaming output: non-temporal b128 stores so the write stream
        // doesn't evict the L2-resident embedding table.
        __builtin_nontemporal_store(o0, &o4[r * 64 + lane]);
        __builtin_nontemporal_store(o1, &o4[r * 64 + lane + 32]);
    }
}

extern "C" void kernel_launch(void* const* d_in, const int* in_sizes, int n_in,
                              void* d_out, int out_size, void* d_ws, size_t ws_size,
                              hipStream_t stream) {
    const unsigned* i1 = (const unsigned*)d_in[0];  // input_one [16,4096] int32
    const unsigned* i2 = (const unsigned*)d_in[1];  // input_two [16,4096] int32
    const float*    w  = (const float*)d_in[2];     // weight [100000,256] f32
    float*          o  = (float*)d_out;             // [16,4096,256] f32

    const int rows   = in_sizes[0];                  // 65536
    const int groups = rows / ROWS_PER_GROUP;        // 8192
    const int blocks = groups / WAVES_PER_BLOCK;     // 4096

    twohot_tdm_kernel<<<blocks, THREADS, 0, stream>>>(i1, i2, w, o);
}